// MODEL_89507118449159
// MI455X (gfx1250) — compile-verified
//
#include <hip/hip_runtime.h>
#include <math.h>

typedef float v2f __attribute__((ext_vector_type(2)));
typedef float v8f __attribute__((ext_vector_type(8)));

#define BS   32
#define SEQ  50
#define ML   16
#define NNODE 15
#define ENC  128
#define GH   32
#define NC   100
#define LH   128
#define LIN_ 167   // NC+1 + 2*GH + 2
#define ROWS (BS*SEQ)          // 1600
#define TREES (BS*SEQ*ML)      // 25600

__device__ __forceinline__ v8f wmma4(v2f a, v2f b, v8f c) {
  // D = A(16x4 f32) * B(4x16 f32) + C(16x16 f32)
  return __builtin_amdgcn_wmma_f32_16x16x4_f32(false, a, false, b, (short)0, c, false, false);
}

__device__ __forceinline__ float sigm(float x) { return 1.f / (1.f + expf(-x)); }

// ---------------------------------------------------------------------------
// Zero init for recurrent states + loss accumulators
__global__ void k_zero(float* p, int n) {
  int i = blockIdx.x * 256 + threadIdx.x;
  if (i < n) p[i] = 0.f;
}

// ---------------------------------------------------------------------------
// Fused: embedding gather -> binary-tree subtree sums -> GEMM (@Wc^T, WMMA f32)
// -> + cnt*bias -> max over 15 nodes  => tree_enc[TREES][128]
// One wave per tree; 8 trees per 256-thread block. Subtree sums staged in LDS.
__global__ __launch_bounds__(256) void k_tree(const int* __restrict__ tok,
                                              const float* __restrict__ emb,
                                              const float* __restrict__ Wc,
                                              const float* __restrict__ Wcb,
                                              float* __restrict__ tree_enc) {
  __shared__ float Slds[8][16 * 128];   // 64 KB, one 16x128 A-tile per wave
  int wave = threadIdx.x >> 5, lane = threadIdx.x & 31;
  int half = lane >> 4, l16 = lane & 15;
  int tree = blockIdx.x * 8 + wave;
  float* S = &Slds[wave][0];
  const int* mytok = tok + tree * NNODE;

  // gather: each lane owns 4 dims (float4) per node
  int d = lane * 4;
#pragma unroll
  for (int n = 0; n < NNODE; ++n) {
    int r = mytok[n];
    *(float4*)(S + n * 128 + d) = *(const float4*)(emb + (size_t)r * 128 + d);
  }
  // bottom-up subtree sums (children indices > parent, finalize high->low)
#pragma unroll
  for (int par = 6; par >= 0; --par) {
    float4 a = *(float4*)(S + (2 * par + 1) * 128 + d);
    float4 b = *(float4*)(S + (2 * par + 2) * 128 + d);
    float4 s = *(float4*)(S + par * 128 + d);
    s.x += a.x + b.x; s.y += a.y + b.y; s.z += a.z + b.z; s.w += a.w + b.w;
    *(float4*)(S + par * 128 + d) = s;
  }
  // dummy row 15 := row 0 (max-neutral)
  *(float4*)(S + 15 * 128 + d) = *(float4*)(S + 0 * 128 + d);

  // GEMM: 16x128 A-tile x 128x128 (Wc^T), f32 WMMA, K=128 in chunks of 4
  v8f acc[8] = {};
  for (int kc = 0; kc < 32; ++kc) {
    int kb = kc * 4 + half * 2;
    v2f a = *(const v2f*)(S + l16 * 128 + kb);
#pragma unroll
    for (int nt = 0; nt < 8; ++nt) {
      v2f b = *(const v2f*)(Wc + (nt * 16 + l16) * 128 + kb);  // B[k][n]=Wc[n][k]
      acc[nt] = wmma4(a, b, acc[nt]);
    }
  }
  // epilogue: + cnt(node)*Wcb[n], max over nodes, combine lane halves
#pragma unroll
  for (int nt = 0; nt < 8; ++nt) {
    float bv = Wcb[nt * 16 + l16];
    float m = -1e30f;
#pragma unroll
    for (int v = 0; v < 8; ++v) {
      int node = v + 8 * half;
      float cnt = (node == 0) ? 15.f : (node < 3 ? 7.f : (node < 7 ? 3.f : (node < 15 ? 1.f : 15.f)));
      m = fmaxf(m, acc[nt][v] + cnt * bv);
    }
    float o = __shfl_xor(m, 16, 32);
    m = fmaxf(m, o);
    if (half == 0) tree_enc[(size_t)tree * 128 + nt * 16 + l16] = m;
  }
}

// ---------------------------------------------------------------------------
// Generic WMMA f32 GEMM: C[M][N] = A[M][K] @ Wt[N][K]^T (+bias[N])
// 8 waves/block, wave = 16 rows, blockIdx.y = 64-column chunk (4 N-tiles).
// Wt chunk staged in LDS (zero-padded to Kp, Kp%4==0).
__global__ __launch_bounds__(256) void k_gemm(const float* __restrict__ A,
                                              const float* __restrict__ Wt,
                                              const float* __restrict__ bias,
                                              float* __restrict__ C,
                                              int M, int N, int K, int Kp) {
  extern __shared__ float Wlds[];  // 64 * Kp
  int n0 = blockIdx.y * 64;
  for (int i = threadIdx.x; i < 64 * Kp; i += 256) {
    int r = i / Kp, c = i % Kp;
    int n = n0 + r;
    Wlds[i] = (n < N && c < K) ? Wt[(size_t)n * K + c] : 0.f;
  }
  __syncthreads();

  int wave = threadIdx.x >> 5, lane = threadIdx.x & 31;
  int half = lane >> 4, l16 = lane & 15;
  int m0 = blockIdx.x * 128 + wave * 16;
  int row = m0 + l16;
  v8f acc[4] = {};
  int nkc = Kp >> 2;
  for (int kc = 0; kc < nkc; ++kc) {
    int kb = kc * 4 + half * 2;
    v2f a;
    a.x = (row < M && kb < K)     ? A[(size_t)row * K + kb]     : 0.f;
    a.y = (row < M && kb + 1 < K) ? A[(size_t)row * K + kb + 1] : 0.f;
#pragma unroll
    for (int nt = 0; nt < 4; ++nt) {
      v2f b = *(const v2f*)(&Wlds[(nt * 16 + l16) * Kp + kb]);
      acc[nt] = wmma4(a, b, acc[nt]);
    }
  }
#pragma unroll
  for (int nt = 0; nt < 4; ++nt) {
    int col = n0 + nt * 16 + l16;
    float bv = (bias != nullptr && col < N) ? bias[col] : 0.f;
#pragma unroll
    for (int v = 0; v < 8; ++v) {
      int r = m0 + v + 8 * half;
      if (r < M && col < N) C[(size_t)r * N + col] = acc[nt][v] + bv;
    }
  }
}

// ---------------------------------------------------------------------------
// One GRU timestep: gh = h @ Wh^T (WMMA) -> LDS -> gates -> h', store hs[tIdx]
// M=1600 rows, K=32, N=96. 4 waves/block, 25 blocks.
__global__ __launch_bounds__(128) void k_gru_step(const float* __restrict__ xe,
                                                  const float* __restrict__ Wh,
                                                  const float* __restrict__ bh,
                                                  float* __restrict__ h,
                                                  float* __restrict__ hs,
                                                  int tIdx) {
  __shared__ float WhL[96 * 32];
  __shared__ float ghL[4][16 * 96];
  for (int i = threadIdx.x; i < 96 * 32; i += 128) WhL[i] = Wh[i];
  __syncthreads();

  int wave = threadIdx.x >> 5, lane = threadIdx.x & 31;
  int half = lane >> 4, l16 = lane & 15;
  int m0 = blockIdx.x * 64 + wave * 16;
  v8f acc[6] = {};
  for (int kc = 0; kc < 8; ++kc) {
    int kb = kc * 4 + half * 2;
    v2f a = *(const v2f*)(h + (m0 + l16) * 32 + kb);
#pragma unroll
    for (int nt = 0; nt < 6; ++nt) {
      v2f b = *(const v2f*)(&WhL[(nt * 16 + l16) * 32 + kb]);
      acc[nt] = wmma4(a, b, acc[nt]);
    }
  }
  float* G = &ghL[wave][0];
#pragma unroll
  for (int nt = 0; nt < 6; ++nt)
#pragma unroll
    for (int v = 0; v < 8; ++v)
      G[(v + 8 * half) * 96 + nt * 16 + l16] = acc[nt][v];
  __syncthreads();

  int c = lane;  // hidden index 0..31
  float bhr = bh[c], bhz = bh[32 + c], bhn = bh[64 + c];
  for (int r = 0; r < 16; ++r) {
    int row = m0 + r;
    const float* xt = xe + ((size_t)row * 16 + tIdx) * 96;
    float ghr = G[r * 96 + c] + bhr;
    float ghz = G[r * 96 + 32 + c] + bhz;
    float ghn = G[r * 96 + 64 + c] + bhn;
    float rr = sigm(xt[c] + ghr);
    float zz = sigm(xt[32 + c] + ghz);
    float nn = tanhf(xt[64 + c] + rr * ghn);
    float hp = h[row * 32 + c];
    float hv = (1.f - zz) * nn + zz * hp;
    h[row * 32 + c] = hv;
    hs[(size_t)row * 512 + tIdx * 32 + c] = hv;
  }
}

// ---------------------------------------------------------------------------
// Build lin[1600][167] = [c_embed(101) | max_t concat(hf,hb)(64) | cur_result(2)]
__global__ void k_lincode(const float* __restrict__ cemb, const float* __restrict__ curr,
                          const float* __restrict__ hsF, const float* __restrict__ hsB,
                          float* __restrict__ lin) {
  int idx = blockIdx.x * 256 + threadIdx.x;
  if (idx >= ROWS * LIN_) return;
  int row = idx / LIN_, j = idx % LIN_;
  float val;
  if (j < NC + 1) {
    val = cemb[(size_t)row * (NC + 1) + j];
  } else if (j < NC + 1 + 2 * GH) {
    int g = j - (NC + 1);
    const float* base = (g < GH) ? (hsF + (size_t)row * 512 + g) : (hsB + (size_t)row * 512 + (g - GH));
    float m = -1e30f;
#pragma unroll
    for (int t = 0; t < ML; ++t) m = fmaxf(m, base[t * 32]);
    val = m;
  } else {
    val = curr[row * 2 + (j - (NC + 1 + 2 * GH))];
  }
  lin[idx] = val;
}

// ---------------------------------------------------------------------------
// One LSTM timestep: g = xe_t + h @ Wh^T + bh (WMMA, Wh from L2) -> LDS ->
// gates -> c', h'; store lout[b*SEQ+t]. Single 256-thread block.
__global__ __launch_bounds__(256) void k_lstm_step(const float* __restrict__ xe,
                                                   const float* __restrict__ Wh,
                                                   const float* __restrict__ bh,
                                                   float* __restrict__ h,
                                                   float* __restrict__ c,
                                                   float* __restrict__ lout,
                                                   int t) {
  __shared__ float G[32 * 512];  // 64 KB gate tile
  int wave = threadIdx.x >> 5, lane = threadIdx.x & 31;
  int half = lane >> 4, l16 = lane & 15;
  int mt = wave >> 2;            // 0..1 -> rows 0..15 / 16..31
  int ntg = (wave & 3) * 8;      // 8 N-tiles per wave
  int m0 = mt * 16;
  v8f acc[8] = {};
  for (int kc = 0; kc < 32; ++kc) {
    int kb = kc * 4 + half * 2;
    v2f a = *(const v2f*)(h + (m0 + l16) * 128 + kb);
#pragma unroll
    for (int nt = 0; nt < 8; ++nt) {
      v2f b = *(const v2f*)(Wh + (size_t)((ntg + nt) * 16 + l16) * 128 + kb);
      acc[nt] = wmma4(a, b, acc[nt]);
    }
  }
#pragma unroll
  for (int nt = 0; nt < 8; ++nt) {
    int col = (ntg + nt) * 16 + l16;
#pragma unroll
    for (int v = 0; v < 8; ++v) {
      int row = m0 + v + 8 * half;
      G[row * 512 + col] = acc[nt][v] + xe[((size_t)row * SEQ + t) * 512 + col] + bh[col];
    }
  }
  __syncthreads();
  for (int idx = threadIdx.x; idx < 32 * 128; idx += 256) {
    int row = idx >> 7, cc = idx & 127;
    float gi = G[row * 512 + cc];
    float gf = G[row * 512 + 128 + cc];
    float gg = G[row * 512 + 256 + cc];
    float go = G[row * 512 + 384 + cc];
    float iv = sigm(gi), fv = sigm(gf), g2 = tanhf(gg), ov = sigm(go);
    float cn = fv * c[idx] + iv * g2;
    float hn = ov * tanhf(cn);
    c[idx] = cn;
    h[idx] = hn;
    lout[((size_t)row * SEQ + t) * 128 + cc] = hn;
  }
}

// ---------------------------------------------------------------------------
// Loss: per-row dot with one-hot target, BCE, global reduction via atomics
__global__ void k_loss1(const float* __restrict__ pred, const float* __restrict__ tc,
                        const float* __restrict__ result, float* __restrict__ out,
                        float* __restrict__ accs) {
  int i = blockIdx.x * 256 + threadIdx.x;
  if (i >= ROWS) return;
  float p1 = 0.f, ncon = 0.f;
  for (int j = 0; j < NC + 1; ++j) {
    float tv = tc[(size_t)i * (NC + 1) + j];
    p1 += pred[(size_t)i * (NC + 1) + j] * tv;
    ncon += tv;
  }
  float m = (ncon > 0.f) ? 1.f : 0.f;
  float fp = p1 / fmaxf(ncon, 1.f);
  float bce = fmaxf(fp, 0.f) - fp * result[i] + log1pf(expf(-fabsf(fp)));
  atomicAdd(&accs[0], bce * m);
  atomicAdd(&accs[1], m);
  out[1 + i] = sigm(fp) * m;
  out[1 + ROWS + i] = result[i] * m;
}

__global__ void k_loss2(const float* __restrict__ accs, float* __restrict__ out) {
  out[0] = accs[0] / fmaxf(accs[1], 1.f);
}

// ---------------------------------------------------------------------------
extern "C" void kernel_launch(void* const* d_in, const int* in_sizes, int n_in,
                              void* d_out, int out_size, void* d_ws, size_t ws_size,
                              hipStream_t stream) {
  const int*   ast  = (const int*)  d_in[2];
  const float* tgt  = (const float*)d_in[3];
  const float* cemb = (const float*)d_in[4];
  const float* curr = (const float*)d_in[5];
  const float* res  = (const float*)d_in[6];
  const float* emb  = (const float*)d_in[7];
  const float* Wc   = (const float*)d_in[8];
  const float* Wcb  = (const float*)d_in[9];
  const float* WxF  = (const float*)d_in[10];
  const float* WhF  = (const float*)d_in[11];
  const float* bxF  = (const float*)d_in[12];
  const float* bhF  = (const float*)d_in[13];
  const float* WxB  = (const float*)d_in[14];
  const float* WhB  = (const float*)d_in[15];
  const float* bxB  = (const float*)d_in[16];
  const float* bhB  = (const float*)d_in[17];
  const float* lWx  = (const float*)d_in[18];
  const float* lWh  = (const float*)d_in[19];
  const float* lbx  = (const float*)d_in[20];
  const float* lbh  = (const float*)d_in[21];
  const float* pw   = (const float*)d_in[22];
  const float* pb   = (const float*)d_in[23];
  float* out = (float*)d_out;

  float* ws = (float*)d_ws;
  size_t o = 0;
  float* tree_enc = ws + o; o += (size_t)TREES * ENC;
  float* xeF      = ws + o; o += (size_t)TREES * 96;
  float* xeB      = ws + o; o += (size_t)TREES * 96;
  float* hsF      = ws + o; o += (size_t)ROWS * ML * GH;
  float* hsB      = ws + o; o += (size_t)ROWS * ML * GH;
  float* lin      = ws + o; o += (size_t)ROWS * LIN_;
  float* lxe      = ws + o; o += (size_t)ROWS * 512;
  float* lout     = ws + o; o += (size_t)ROWS * LH;
  float* pred     = ws + o; o += (size_t)ROWS * (NC + 1);
  float* zbase    = ws + o;   // contiguous zero region below
  float* hF       = ws + o; o += (size_t)ROWS * GH;
  float* hB       = ws + o; o += (size_t)ROWS * GH;
  float* lh       = ws + o; o += 32 * LH;
  float* lc       = ws + o; o += 32 * LH;
  float* accs     = ws + o; o += 2;
  int nzero = ROWS * GH * 2 + 32 * LH * 2 + 2;

  // 0) zero recurrent states + accumulators
  k_zero<<<(nzero + 255) / 256, 256, 0, stream>>>(zbase, nzero);

  // 1) fused gather + tree-sum + WMMA GEMM + node-max
  k_tree<<<TREES / 8, 256, 0, stream>>>(ast, emb, Wc, Wcb, tree_enc);

  // 2) GRU input projections: xe = tree_enc @ Wx^T + bx  (M=25600,K=128,N=96)
  k_gemm<<<dim3(TREES / 128, 2), 256, 64 * 128 * 4, stream>>>(tree_enc, WxF, bxF, xeF, TREES, 96, 128, 128);
  k_gemm<<<dim3(TREES / 128, 2), 256, 64 * 128 * 4, stream>>>(tree_enc, WxB, bxB, xeB, TREES, 96, 128, 128);

  // 3) GRU recurrences (forward & reversed time)
  for (int t = 0; t < ML; ++t)
    k_gru_step<<<25, 128, 0, stream>>>(xeF, WhF, bhF, hF, hsF, t);
  for (int t = 0; t < ML; ++t)
    k_gru_step<<<25, 128, 0, stream>>>(xeB, WhB, bhB, hB, hsB, ML - 1 - t);

  // 4) code = max_t concat(hf,hb); lin = [c_embed | code | cur_result]
  k_lincode<<<(ROWS * LIN_ + 255) / 256, 256, 0, stream>>>(cemb, curr, hsF, hsB, lin);

  // 5) LSTM input projection: lxe = lin @ lstm_Wx^T + bx (M=1600,K=167,N=512)
  k_gemm<<<dim3((ROWS + 127) / 128, 8), 256, 64 * 168 * 4, stream>>>(lin, lWx, lbx, lxe, ROWS, 512, 167, 168);

  // 6) LSTM recurrence (50 steps)
  for (int t = 0; t < SEQ; ++t)
    k_lstm_step<<<1, 256, 0, stream>>>(lxe, lWh, lbh, lh, lc, lout, t);

  // 7) pred = lout @ pred_w^T + pred_b (M=1600,K=128,N=101)
  k_gemm<<<dim3((ROWS + 127) / 128, 2), 256, 64 * 128 * 4, stream>>>(lout, pw, pb, pred, ROWS, NC + 1, 128, 128);

  // 8) loss
  k_loss1<<<(ROWS + 255) / 256, 256, 0, stream>>>(pred, tgt, res, out, accs);
  k_loss2<<<1, 1, 0, stream>>>(accs, out);
}